// HOSVDAggregator_84293028151828
// MI455X (gfx1250) — compile-verified
//
#include <hip/hip_runtime.h>

typedef __attribute__((ext_vector_type(2))) float v2f;
typedef __attribute__((ext_vector_type(4))) float v4f;
typedef __attribute__((ext_vector_type(8))) float v8f;

#define WMMA_F32(a, b, c) \
  __builtin_amdgcn_wmma_f32_16x16x4_f32(false, (a), false, (b), (short)0, (c), false, false)

#define NN 50000
#define TILES 3125            // NN / 16
#define WPB 4                 // waves per block

// d_ws layout (floats):
//   Gf : [128 ksteps][32 lanes][4 nc][2 j]  = 32768 floats (128 KB)  B-fragments of G.reshape(64,512)^T
//   Uf : [4 d][64 ksteps][32 lanes][2 j]    = 16384 floats (64 KB)   B-fragments of U_stack (cols 8..15 = 0)

__global__ void hosvd_prep(const float* __restrict__ G,
                           const float* __restrict__ Us,
                           float* __restrict__ ws) {
  int idx = blockIdx.x * blockDim.x + threadIdx.x;
  if (idx < 32768) {
    // Gf[((ks*32 + l)*4 + nc)*2 + j] = B[k = ks*4 + 2*(l>>4) + j, n = nc*16 + (l&15)] = Gflat[n*512 + k]
    int j  = idx & 1;
    int nc = (idx >> 1) & 3;
    int l  = (idx >> 3) & 31;
    int ks = idx >> 8;
    int n  = nc * 16 + (l & 15);
    int k  = ks * 4 + 2 * (l >> 4) + j;
    ws[idx] = G[n * 512 + k];
  } else if (idx < 49152) {
    int u  = idx - 32768;
    int j  = u & 1;
    int l  = (u >> 1) & 31;
    int ks = (u >> 6) & 63;
    int d  = u >> 12;
    int n  = l & 15;
    int k  = ks * 4 + 2 * (l >> 4) + j;
    ws[idx] = (n < 8) ? Us[d * 2048 + k * 8 + n] : 0.0f;
  }
}

__global__ __launch_bounds__(128) void hosvd_main(const float* __restrict__ ns,
                                                  const float* __restrict__ Uo,
                                                  const float* __restrict__ ws,
                                                  float* __restrict__ out) {
  __shared__ float ldsUo[1024];          // U_output [8][128]
  __shared__ float ldsRis[WPB][16 * 36]; // per wave: 16 nodes x (4 slots * 8 ranks), stride 36 (bank pad)
  __shared__ float ldsT[WPB][16 * 65];   // per wave: 16 nodes x 64 (r0,r1), stride 65 (bank pad)

  const float* Gf = ws;
  const float* Uf = ws + 32768;

  for (int i = threadIdx.x; i < 1024; i += 128) ldsUo[i] = Uo[i];
  __syncthreads();

  const int wave = threadIdx.x >> 5;
  const int lane = threadIdx.x & 31;
  const int m    = lane & 15;   // node-in-tile (A rows) / B column
  const int hi   = lane >> 4;   // K half-select

  const int tile = blockIdx.x * WPB + wave;
  if (tile >= TILES) return;    // wave-uniform
  const int nodeBase = tile * 16;

  float* ris = &ldsRis[wave][0];
  float* T   = &ldsT[wave][0];

  // ---------------- Step 1: projection ris[n,d,r] = state[n,d,:] @ U_stack[d] ----------------
  const float* aBase = ns + (size_t)(nodeBase + m) * 1024 + hi * 2;
  for (int d = 0; d < 4; ++d) {
    v8f acc{};
    const float* ap = aBase + d * 256;
    const float* bp = Uf + (size_t)d * 4096;
#pragma unroll 8
    for (int ks = 0; ks < 64; ++ks) {
      v2f a = *(const v2f*)(ap + ks * 4);           // A[m, ks*4 + 2*hi + {0,1}]
      v2f b = *(const v2f*)(bp + ks * 64 + lane * 2);
      acc = WMMA_F32(a, b, acc);
    }
    if (m < 8) {                                    // cols 0..7 hold ris; write to LDS
#pragma unroll
      for (int v = 0; v < 8; ++v)
        ris[(v + 8 * hi) * 36 + d * 8 + m] = acc[v];
    }
  }
  asm volatile("s_wait_dscnt 0" ::: "memory");

  // per-lane ris values for on-the-fly w = ris3[r2]*ris2[r3]*ris1[r4] A-fragments
  float a00 = ris[m * 36 + 8 + 2 * hi];
  float a01 = ris[m * 36 + 8 + 2 * hi + 1];
  float a10 = ris[m * 36 + 8 + 2 * hi + 4];
  float a11 = ris[m * 36 + 8 + 2 * hi + 5];
  float r2v[8], r3v[8];
#pragma unroll
  for (int r = 0; r < 8; ++r) {
    r2v[r] = ris[m * 36 + 16 + r];
    r3v[r] = ris[m * 36 + 24 + r];
  }

  // ---------------- Steps 2+3: T[16 x 64] = w[16 x 512] @ G2^T[512 x 64] ----------------
  v8f c0{}, c1{}, c2{}, c3{};
  const float* gb = Gf + lane * 8;
#pragma unroll
  for (int ks = 0; ks < 128; ++ks) {
    // A-fragment generated in registers (compile-time indices after full unroll)
    float t = r3v[ks >> 4] * r2v[(ks >> 1) & 7];
    v2f a;
    if (ks & 1) { a.x = t * a10; a.y = t * a11; }
    else        { a.x = t * a00; a.y = t * a01; }
    const v4f* bp = (const v4f*)(gb + ks * 256);    // coalesced, L2-resident
    v4f b01 = bp[0], b23 = bp[1];
    v2f b0 = {b01.x, b01.y}, b1 = {b01.z, b01.w};
    v2f b2 = {b23.x, b23.y}, b3 = {b23.z, b23.w};
    c0 = WMMA_F32(a, b0, c0);
    c1 = WMMA_F32(a, b1, c1);
    c2 = WMMA_F32(a, b2, c2);
    c3 = WMMA_F32(a, b3, c3);
  }

  // write T[node][(r0,r1)] to LDS
#pragma unroll
  for (int v = 0; v < 8; ++v) {
    int base = (v + 8 * hi) * 65 + m;
    T[base]      = c0[v];
    T[base + 16] = c1[v];
    T[base + 32] = c2[v];
    T[base + 48] = c3[v];
  }
  asm volatile("s_wait_dscnt 0" ::: "memory");

  // ---------------- Steps 4+5: s[r1] = sum_r0 ris0[r0]*T[r0*8+r1];  out = s @ U_out ----------------
  float s[8];
#pragma unroll
  for (int r1 = 0; r1 < 8; ++r1) {
    float acc = 0.0f;
#pragma unroll
    for (int r0 = 0; r0 < 8; ++r0)
      acc += ris[m * 36 + r0] * T[m * 65 + r0 * 8 + r1];
    s[r1] = acc;
  }
  float* op = out + (size_t)(nodeBase + m) * 128 + hi * 64;
#pragma unroll
  for (int o4 = 0; o4 < 16; ++o4) {
    v4f acc{};
#pragma unroll
    for (int r1 = 0; r1 < 8; ++r1) {
      v4f u = *(const v4f*)(ldsUo + r1 * 128 + hi * 64 + o4 * 4);
      acc.x += s[r1] * u.x;
      acc.y += s[r1] * u.y;
      acc.z += s[r1] * u.z;
      acc.w += s[r1] * u.w;
    }
    *(v4f*)(op + o4 * 4) = acc;
  }
}

extern "C" void kernel_launch(void* const* d_in, const int* in_sizes, int n_in,
                              void* d_out, int out_size, void* d_ws, size_t ws_size,
                              hipStream_t stream) {
  const float* ns = (const float*)d_in[0];  // [50000,4,256]
  const float* G  = (const float*)d_in[1];  // [8,8,8,8,8]
  const float* Us = (const float*)d_in[2];  // [4,256,8]
  const float* Uo = (const float*)d_in[3];  // [8,128]
  float* ws  = (float*)d_ws;                // needs >= 192 KB
  float* out = (float*)d_out;               // [50000,128]

  hipLaunchKernelGGL(hosvd_prep, dim3(192), dim3(256), 0, stream, G, Us, ws);
  hipLaunchKernelGGL(hosvd_main, dim3((TILES + WPB - 1) / WPB), dim3(128), 0, stream,
                     ns, Uo, (const float*)ws, out);
}